// LorentzMultiHeadAttention_21199958573249
// MI455X (gfx1250) — compile-verified
//
#include <hip/hip_runtime.h>
#include <math.h>

typedef __attribute__((ext_vector_type(2))) float v2f;
typedef __attribute__((ext_vector_type(8))) float v8f;

#define NHEADS   12
#define HD       64
#define NTOK     1024
#define BATCH    8
#define FDIM     769
#define BHDIM    (BATCH * NHEADS)
#define VSTRIDE  80            /* v rows padded: [0]=time, [1..64]=space, [65..79]=0 */
#define EPSF     1e-6f

static __device__ inline v8f wmma_f32(v2f a, v2f b, v8f c) {
    // V_WMMA_F32_16X16X4_F32 : D = A(16x4) * B(4x16) + C(16x16), wave32
    return __builtin_amdgcn_wmma_f32_16x16x4_f32(
        /*neg_a=*/false, a, /*neg_b=*/false, b,
        /*c_mod=*/(short)0, c, /*reuse_a=*/false, /*reuse_b=*/false);
}

// ---------------------------------------------------------------------------
// GEMM: out_space = X(8192x769) @ W[:,1:769]  (space columns only, N=768)
// mode 0: write head-split layout  [(b*12+h)*1024+n]*64 + d           (q/k space)
// mode 1: write padded V layout    [(b*12+h)*1024+n]*80 + 1 + d
// mode 2: write row-major          [row]*769 + 1 + c                  (final proj)
// Block = 128 threads (4 waves); wave computes a 16(M) x 64(N) tile.
// ---------------------------------------------------------------------------
__global__ __launch_bounds__(128) void gemm_space_kernel(
    const float* __restrict__ X, const float* __restrict__ W,
    float* __restrict__ out, int mode)
{
    const int wave = threadIdx.x >> 5;
    const int lane = threadIdx.x & 31;
    const int lm   = lane & 15;
    const int hi   = (lane >> 4) & 1;   // 0: K=0,1   1: K=2,3
    const int koff = hi * 2;
    const int m0   = blockIdx.x * 64 + wave * 16;
    const int n0   = blockIdx.y * 64;
    const float kmask = hi ? 0.0f : 1.0f;   // branch-free K-remainder masking

    v8f acc[4];
    for (int t = 0; t < 4; ++t) acc[t] = (v8f){};

    const float* xrow = X + (size_t)(m0 + lm) * FDIM;

    for (int k0 = 0; k0 < 768; k0 += 4) {
        __builtin_prefetch(xrow + k0 + 96, 0, 1);   // global_prefetch_b8
        v2f a;
        a.x = xrow[k0 + koff];
        a.y = xrow[k0 + koff + 1];
        const float* wr0 = W + (size_t)(k0 + koff) * FDIM + 1 + n0;
        const float* wr1 = wr0 + FDIM;
        for (int t = 0; t < 4; ++t) {
            v2f b;
            b.x = wr0[t * 16 + lm];
            b.y = wr1[t * 16 + lm];
            acc[t] = wmma_f32(a, b, acc[t]);
        }
    }
    { // K remainder: k = 768 (K=769). Unconditional in-bounds loads, then a
      // v_cndmask-style select zeroes the lanes that would cover k>=769.
        v2f a;
        a.x = xrow[768] * kmask;
        a.y = 0.0f;
        const float* wr0 = W + (size_t)768 * FDIM + 1 + n0;
        for (int t = 0; t < 4; ++t) {
            v2f b;
            b.x = wr0[t * 16 + lm] * kmask;
            b.y = 0.0f;
            acc[t] = wmma_f32(a, b, acc[t]);
        }
    }

    // C/D layout: vgpr r, lanes0-15 -> M=r, lanes16-31 -> M=r+8; N=lane&15
    for (int t = 0; t < 4; ++t) {
        const int c = n0 + t * 16 + lm;          // space column 0..767
        for (int r = 0; r < 8; ++r) {
            const int row = m0 + r + hi * 8;     // global token row 0..8191
            const float val = acc[t][r];
            if (mode == 2) {
                out[(size_t)row * FDIM + 1 + c] = val;
            } else {
                const int bb = row >> 10, nn = row & 1023;
                const int h = c >> 6, d = c & 63;
                const size_t tok = (size_t)(bb * NHEADS + h) * NTOK + nn;
                if (mode == 0) out[tok * HD + d] = val;
                else           out[tok * VSTRIDE + 1 + d] = val;
            }
        }
    }
}

// time = sqrt(1 + ||space||^2) for q/k rows (head-split, stride 64)
__global__ void qk_time_kernel(const float* __restrict__ s, float* __restrict__ t, int rows)
{
    const int r = blockIdx.x * blockDim.x + threadIdx.x;
    if (r >= rows) return;
    const float* p = s + (size_t)r * HD;
    float acc = 0.0f;
    for (int d = 0; d < HD; ++d) acc = fmaf(p[d], p[d], acc);
    t[r] = sqrtf(1.0f + acc);
}

// v row: compute v[0]=time, zero pad v[65..79]
__global__ void v_time_pad_kernel(float* __restrict__ v, int rows)
{
    const int r = blockIdx.x * blockDim.x + threadIdx.x;
    if (r >= rows) return;
    float* p = v + (size_t)r * VSTRIDE;
    float acc = 0.0f;
    for (int d = 1; d <= HD; ++d) acc = fmaf(p[d], p[d], acc);
    p[0] = sqrtf(1.0f + acc);
    for (int d = HD + 1; d < VSTRIDE; ++d) p[d] = 0.0f;
}

// ---------------------------------------------------------------------------
// Fused Lorentz attention per (b,h). For output token j:
//   m[j,:] = sum_i exp( 1/(1+log1p(max(-2-2*inner(i,j),eps))) ) * v[i,:]
//   attn[j,:] = m / sqrt(max(|  ||m_s||^2 - m_t^2 |, eps))     (Z cancels)
// Logits are in (0,1] -> exp is bounded, no online max needed.
// Block = 128 threads; wave w owns j-rows [j0+16w, j0+16w+16).
// ---------------------------------------------------------------------------
__global__ __launch_bounds__(128) void lorentz_attn_kernel(
    const float* __restrict__ qs, const float* __restrict__ qt,
    const float* __restrict__ ks, const float* __restrict__ kt,
    const float* __restrict__ v,  float* __restrict__ att)
{
    __shared__ float ldsE[4][16 * 16];   // per-wave E-tile transpose buffer

    const int wave = threadIdx.x >> 5;
    const int lane = threadIdx.x & 31;
    const int lm   = lane & 15;
    const int hi   = (lane >> 4) & 1;
    const int koff = hi * 2;
    const int bh   = blockIdx.x;
    const int j0   = blockIdx.y * 64 + wave * 16;

    const float* qs_b = qs + (size_t)bh * NTOK * HD;
    const float* ks_b = ks + (size_t)bh * NTOK * HD;
    const float* v_b  = v  + (size_t)bh * NTOK * VSTRIDE;
    const float* qt_b = qt + (size_t)bh * NTOK;
    const float* kt_b = kt + (size_t)bh * NTOK;
    float* myE = &ldsE[wave][0];

    // Hoist A-side (K-proj rows j) WMMA fragments for the whole i-loop.
    v2f afr[16];
    const float* krow = ks_b + (size_t)(j0 + lm) * HD;
    for (int s = 0; s < 16; ++s) {
        afr[s].x = krow[4 * s + koff];
        afr[s].y = krow[4 * s + koff + 1];
    }
    float ktv[8];
    for (int r = 0; r < 8; ++r) ktv[r] = kt_b[j0 + r + hi * 8];

    v8f macc[5];
    for (int t = 0; t < 5; ++t) macc[t] = (v8f){};

    for (int i0 = 0; i0 < NTOK; i0 += 16) {
        // S^T(j,i) = sum_d ks[j,d]*qs[i,d]  : 16 WMMA k-steps over d
        v8f sacc = (v8f){};
        const float* qrow = qs_b + (size_t)(i0 + lm) * HD;
        for (int s = 0; s < 16; ++s) {
            v2f b;
            b.x = qrow[4 * s + koff];
            b.y = qrow[4 * s + koff + 1];
            sacc = wmma_f32(afr[s], b, sacc);
        }
        // logits -> unnormalized exp weights, staged to LDS row-major [j][i]
        const float qtl = qt_b[i0 + lm];
        for (int r = 0; r < 8; ++r) {
            const float inner = sacc[r] - ktv[r] * qtl;          // Lorentz inner
            const float d2 = fmaxf(-2.0f - 2.0f * inner, EPSF);
            const float sc = 1.0f / (1.0f + log1pf(d2));
            myE[(r + hi * 8) * 16 + lm] = __expf(sc);
        }
        asm volatile("s_wait_dscnt 0" ::: "memory");  // same-wave LDS RAW

        // m += E(16j x 16i) * V(16i x 80dd)  : 4 k-steps x 5 N-tiles
        for (int kk = 0; kk < 4; ++kk) {
            v2f ea;
            ea.x = myE[lm * 16 + 4 * kk + koff];
            ea.y = myE[lm * 16 + 4 * kk + koff + 1];
            const float* vr = v_b + (size_t)(i0 + 4 * kk + koff) * VSTRIDE;
            for (int T = 0; T < 5; ++T) {
                v2f vb;
                vb.x = vr[T * 16 + lm];
                vb.y = vr[VSTRIDE + T * 16 + lm];
                macc[T] = wmma_f32(ea, vb, macc[T]);
            }
        }
    }

    // Lorentz centroid normalization per row j (16-lane half-wave reduction).
    float inv[8];
    for (int r = 0; r < 8; ++r) {
        float p = 0.0f;
        for (int T = 0; T < 5; ++T) {
            const float m = macc[T][r];
            float sq = m * m;
            if (T == 0 && lm == 0) sq = -sq;   // dd==0 is the time coord
            p += sq;                           // dd in [65,80) is exactly 0
        }
        p += __shfl_xor(p, 1, 32);
        p += __shfl_xor(p, 2, 32);
        p += __shfl_xor(p, 4, 32);
        p += __shfl_xor(p, 8, 32);             // stays within 16-lane half
        inv[r] = rsqrtf(fmaxf(fabsf(p), EPSF));
    }
    for (int T = 0; T < 5; ++T) {
        const int dd = T * 16 + lm;
        for (int r = 0; r < 8; ++r) {
            if (dd < HD + 1) {
                const int j = j0 + r + hi * 8;
                att[((size_t)bh * NTOK + j) * (HD + 1) + dd] = macc[T][r] * inv[r];
            }
        }
    }
}

// merged(b,n,769): col0 = sqrt(max(sum_h time_h^2 - 11, eps)); cols 1.. = head spaces
__global__ void merge_kernel(const float* __restrict__ att, float* __restrict__ merged)
{
    const size_t idx = (size_t)blockIdx.x * blockDim.x + threadIdx.x;
    const size_t total = (size_t)BATCH * NTOK * FDIM;
    if (idx >= total) return;
    const int r = (int)(idx / FDIM);
    const int c = (int)(idx % FDIM);
    const int bb = r >> 10, nn = r & 1023;
    if (c == 0) {
        float s = 0.0f;
        for (int h = 0; h < NHEADS; ++h) {
            const float t = att[((size_t)(bb * NHEADS + h) * NTOK + nn) * (HD + 1)];
            s = fmaf(t, t, s);
        }
        merged[idx] = sqrtf(fmaxf(s - (float)(NHEADS - 1), EPSF));
    } else {
        const int h = (c - 1) >> 6, d = (c - 1) & 63;
        merged[idx] = att[((size_t)(bb * NHEADS + h) * NTOK + nn) * (HD + 1) + 1 + d];
    }
}

// final add_time on d_out rows (cols 1..768 already written by mode-2 GEMM)
__global__ void out_time_kernel(float* __restrict__ out)
{
    const int r = blockIdx.x * blockDim.x + threadIdx.x;
    if (r >= BATCH * NTOK) return;
    float* p = out + (size_t)r * FDIM;
    float acc = 0.0f;
    for (int c = 1; c < FDIM; ++c) acc = fmaf(p[c], p[c], acc);
    p[0] = sqrtf(1.0f + acc);
}

extern "C" void kernel_launch(void* const* d_in, const int* in_sizes, int n_in,
                              void* d_out, int out_size, void* d_ws, size_t ws_size,
                              hipStream_t stream)
{
    (void)in_sizes; (void)n_in; (void)out_size; (void)ws_size;
    const float* x  = (const float*)d_in[0];
    const float* Wq = (const float*)d_in[1];
    const float* Wk = (const float*)d_in[2];
    const float* Wv = (const float*)d_in[3];
    const float* Wo = (const float*)d_in[4];
    float* out = (float*)d_out;
    float* ws  = (float*)d_ws;

    const size_t rows = (size_t)BHDIM * NTOK;          // 98304
    size_t o = 0;
    float* qs     = ws + o; o += rows * HD;            // 6.29M
    float* ksb    = ws + o; o += rows * HD;            // 6.29M
    float* qt     = ws + o; o += rows;
    float* kt     = ws + o; o += rows;
    float* vbuf   = ws + o; o += rows * VSTRIDE;       // 7.86M
    float* att    = ws + o; o += rows * (HD + 1);      // 6.39M
    float* merged = ws + o; o += (size_t)BATCH * NTOK * FDIM;

    const dim3 gg(8192 / 64, 768 / 64);                // (128, 12)

    gemm_space_kernel<<<gg, 128, 0, stream>>>(x, Wq, qs,   0);
    gemm_space_kernel<<<gg, 128, 0, stream>>>(x, Wk, ksb,  0);
    gemm_space_kernel<<<gg, 128, 0, stream>>>(x, Wv, vbuf, 1);

    const int nrows = (int)rows;
    qk_time_kernel<<<(nrows + 255) / 256, 256, 0, stream>>>(qs,  qt, nrows);
    qk_time_kernel<<<(nrows + 255) / 256, 256, 0, stream>>>(ksb, kt, nrows);
    v_time_pad_kernel<<<(nrows + 255) / 256, 256, 0, stream>>>(vbuf, nrows);

    lorentz_attn_kernel<<<dim3(BHDIM, NTOK / 64), 128, 0, stream>>>(
        qs, qt, ksb, kt, vbuf, att);

    const size_t total = (size_t)BATCH * NTOK * FDIM;
    merge_kernel<<<(unsigned)((total + 255) / 256), 256, 0, stream>>>(att, merged);

    gemm_space_kernel<<<gg, 128, 0, stream>>>(merged, Wo, out, 2);
    out_time_kernel<<<(BATCH * NTOK + 255) / 256, 256, 0, stream>>>(out);
}